// Global_information_61718680044015
// MI455X (gfx1250) — compile-verified
//
#include <hip/hip_runtime.h>

#define DIM 64
#define LTOT 66836
#define BATCH 4

typedef float v2f __attribute__((ext_vector_type(2)));
typedef float v8f __attribute__((ext_vector_type(8)));

// ---------------------------------------------------------------------------
// V_WMMA_F32_16X16X4_F32 : D[16x16] += A[16x4] * B[4x16], all fp32, wave32.
// A: lanes 0-15 hold M=lane with K={k0,k0+1} in {.x,.y};
//    lanes 16-31 hold M=lane-16 with K={k0+2,k0+3}.
// B: mirrored (row K by half, col N = lane&15).
// C/D: vgpr r -> row (r + 8*half), col = n0 + (lane&15).
// ---------------------------------------------------------------------------
__device__ __forceinline__ v8f wmma4(v2f a, v2f b, v8f c) {
  return __builtin_amdgcn_wmma_f32_16x16x4_f32(false, a, false, b, (short)0, c,
                                               false, false);
}

// 16x64 row tile (A preloaded, 16 x v2f per lane) times 64x64 row-major W.
// Produces 4 accumulator tiles covering N = 0..63.
__device__ __forceinline__ void gemm64(const v2f* a, const float* __restrict__ W,
                                       int hx, int lr, v8f acc[4]) {
  v8f z = {};
#pragma unroll
  for (int t = 0; t < 4; t++) acc[t] = z;
  const float* Wp = W + 2 * hx * 64 + lr;  // + s*256 (+64 for .y) + t*16
#pragma unroll
  for (int s = 0; s < 16; s++) {
#pragma unroll
    for (int t = 0; t < 4; t++) {
      v2f bv;
      bv.x = Wp[s * 256 + t * 16];
      bv.y = Wp[s * 256 + 64 + t * 16];
      acc[t] = wmma4(a[s], bv, acc[t]);
    }
  }
}

__global__ void zero_kernel(float* p, int n) {
  int i = blockIdx.x * blockDim.x + threadIdx.x;
  if (i < n) p[i] = 0.f;
}

// ---------------------------------------------------------------------------
// conv1x1 in: x[B,Cin,hw2] -> tokens t1[B,L,64] (raw, bias added, pre-LN)
// ---------------------------------------------------------------------------
__global__ void embed_kernel(const float* __restrict__ x, const float* __restrict__ w,
                             const float* __restrict__ bias, float* __restrict__ t1,
                             int Cin, int hw2, int segStart, int tilesPerBatch) {
  int wid = blockIdx.x * (blockDim.x >> 5) + (threadIdx.x >> 5);
  int b = wid / tilesPerBatch;
  if (b >= BATCH) return;
  int tile = wid - b * tilesPerBatch;
  int lane = threadIdx.x & 31, hx = lane >> 4, lr = lane & 15;
  int pixBase = tile * 16;
  int pix0 = pixBase + lr;
  float mask = pix0 < hw2 ? 1.f : 0.f;
  int pixC = pix0 < hw2 ? pix0 : (hw2 - 1);
  // branchless clamped loads, pointer-stepped (no per-iter 64-bit mul)
  const float* xp = x + (size_t)b * Cin * hw2 + (size_t)(2 * hx) * hw2 + pixC;
  const float* wp[4];
#pragma unroll
  for (int t = 0; t < 4; t++) wp[t] = w + (size_t)(t * 16 + lr) * Cin + 2 * hx;
  v8f z = {};
  v8f acc[4];
#pragma unroll
  for (int t = 0; t < 4; t++) acc[t] = z;
  int steps = Cin >> 2;
  for (int s = 0; s < steps; s++) {
    v2f a;
    a.x = xp[0] * mask;
    a.y = xp[hw2] * mask;
    xp += (size_t)4 * hw2;
#pragma unroll
    for (int t = 0; t < 4; t++) {
      v2f bv = *(const v2f*)wp[t];
      wp[t] += 4;
      acc[t] = wmma4(a, bv, acc[t]);
    }
  }
#pragma unroll
  for (int t = 0; t < 4; t++) {
    int n = t * 16 + lr;
    float bn = bias[n];
#pragma unroll
    for (int r = 0; r < 8; r++) {
      int pix = pixBase + r + 8 * hx;
      if (pix < hw2)
        t1[((size_t)b * LTOT + segStart + pix) * DIM + n] = acc[t][r] + bn;
    }
  }
}

// LayerNorm over the 64-channel axis, one wave per token, in place.
__global__ void ln_tokens_kernel(float* __restrict__ t, const float* __restrict__ g,
                                 const float* __restrict__ bb) {
  int wid = blockIdx.x * (blockDim.x >> 5) + (threadIdx.x >> 5);
  if (wid >= BATCH * LTOT) return;
  int lane = threadIdx.x & 31;
  float* row = t + (size_t)wid * DIM;
  int c0 = 2 * lane;
  v2f v = *(const v2f*)(row + c0);
  float s = v.x + v.y, sq = v.x * v.x + v.y * v.y;
  for (int m = 16; m >= 1; m >>= 1) {
    s += __shfl_xor(s, m, 32);
    sq += __shfl_xor(sq, m, 32);
  }
  float mean = s * (1.f / 64.f);
  float rstd = rsqrtf(sq * (1.f / 64.f) - mean * mean + 1e-5f);
  v2f gv = *(const v2f*)(g + c0);
  v2f bv = *(const v2f*)(bb + c0);
  v2f o;
  o.x = (v.x - mean) * rstd * gv.x + bv.x;
  o.y = (v.y - mean) * rstd * gv.y + bv.y;
  *(v2f*)(row + c0) = o;
}

// ---------------------------------------------------------------------------
// Fused projections: xe = t1 + pos.  Computes Q (stored, relu/8), Q1,K1 raw
// (stored), and streaming stats: Ksum, diag(K^T V), sum/sumsq of Q1,K1 over L.
// stats layout: [6][BATCH][64] = Ksum, KVdiag, q1sum, q1sq, k1sum, k1sq.
// ---------------------------------------------------------------------------
__global__ void proj_kernel(const float* __restrict__ t1, const float* __restrict__ pos,
                            const float* __restrict__ Wq, const float* __restrict__ bq,
                            const float* __restrict__ Wk, const float* __restrict__ bk,
                            const float* __restrict__ Wv, const float* __restrict__ bvp,
                            const float* __restrict__ Wq1, const float* __restrict__ bq1,
                            const float* __restrict__ Wk1, const float* __restrict__ bk1,
                            float* __restrict__ Qout, float* __restrict__ Q1out,
                            float* __restrict__ K1out, float* __restrict__ stats,
                            int tilesPerBatch) {
  int wid = blockIdx.x * (blockDim.x >> 5) + (threadIdx.x >> 5);
  int b = wid / tilesPerBatch;
  if (b >= BATCH) return;
  int tile = wid - b * tilesPerBatch;
  int tok0 = tile * 16;
  int lane = threadIdx.x & 31, hx = lane >> 4, lr = lane & 15;
  int tok = tok0 + lr;
  float mask = tok < LTOT ? 1.f : 0.f;
  int tokC = tok < LTOT ? tok : (LTOT - 1);
  const float* xrow = t1 + ((size_t)b * LTOT + tokC) * DIM + 2 * hx;
  const float* prow = pos + (size_t)tokC * DIM + 2 * hx;
  v2f a[16];
#pragma unroll
  for (int s = 0; s < 16; s++) {
    v2f xv = *(const v2f*)(xrow + 4 * s);
    v2f pv = *(const v2f*)(prow + 4 * s);
    a[s] = (xv + pv) * mask;
  }
  const int B64 = BATCH * 64;
  // ---- V and K together (diag(K^T V) + Ksum) ----
  {
    v8f accV[4], accK[4];
    gemm64(a, Wv, hx, lr, accV);
    gemm64(a, Wk, hx, lr, accK);
#pragma unroll
    for (int t = 0; t < 4; t++) {
      int c = t * 16 + lr;
      float bkc = bk[c], bvc = bvp[c];
      float sK = 0.f, sKV = 0.f;
#pragma unroll
      for (int r = 0; r < 8; r++) {
        int m = r + 8 * hx;
        float rm = (tok0 + m) < LTOT ? 1.f : 0.f;
        float kv = fmaxf(accK[t][r] + bkc, 0.f) * 0.125f * rm;
        float vv = accV[t][r] + bvc;
        sK += kv;
        sKV += kv * vv;
      }
      sK += __shfl_xor(sK, 16, 32);
      sKV += __shfl_xor(sKV, 16, 32);
      if (hx == 0) {
        atomicAdd(&stats[0 * B64 + b * 64 + c], sK);
        atomicAdd(&stats[1 * B64 + b * 64 + c], sKV);
      }
    }
  }
  // ---- Q ----
  {
    v8f acc[4];
    gemm64(a, Wq, hx, lr, acc);
#pragma unroll
    for (int t = 0; t < 4; t++) {
      int c = t * 16 + lr;
      float bc = bq[c];
#pragma unroll
      for (int r = 0; r < 8; r++) {
        int m = r + 8 * hx;
        if ((tok0 + m) < LTOT)
          Qout[((size_t)b * LTOT + tok0 + m) * DIM + c] =
              fmaxf(acc[t][r] + bc, 0.f) * 0.125f;
      }
    }
  }
  // ---- Q1 (raw, + stats for LN over L) ----
  {
    v8f acc[4];
    gemm64(a, Wq1, hx, lr, acc);
#pragma unroll
    for (int t = 0; t < 4; t++) {
      int c = t * 16 + lr;
      float bc = bq1[c];
      float s = 0.f, sq = 0.f;
#pragma unroll
      for (int r = 0; r < 8; r++) {
        int m = r + 8 * hx;
        float v = acc[t][r] + bc;
        if ((tok0 + m) < LTOT) {
          Q1out[((size_t)b * LTOT + tok0 + m) * DIM + c] = v;
          s += v;
          sq += v * v;
        }
      }
      s += __shfl_xor(s, 16, 32);
      sq += __shfl_xor(sq, 16, 32);
      if (hx == 0) {
        atomicAdd(&stats[2 * B64 + b * 64 + c], s);
        atomicAdd(&stats[3 * B64 + b * 64 + c], sq);
      }
    }
  }
  // ---- K1 (raw, + stats) ----
  {
    v8f acc[4];
    gemm64(a, Wk1, hx, lr, acc);
#pragma unroll
    for (int t = 0; t < 4; t++) {
      int c = t * 16 + lr;
      float bc = bk1[c];
      float s = 0.f, sq = 0.f;
#pragma unroll
      for (int r = 0; r < 8; r++) {
        int m = r + 8 * hx;
        float v = acc[t][r] + bc;
        if ((tok0 + m) < LTOT) {
          K1out[((size_t)b * LTOT + tok0 + m) * DIM + c] = v;
          s += v;
          sq += v * v;
        }
      }
      s += __shfl_xor(s, 16, 32);
      sq += __shfl_xor(sq, 16, 32);
      if (hx == 0) {
        atomicAdd(&stats[4 * B64 + b * 64 + c], s);
        atomicAdd(&stats[5 * B64 + b * 64 + c], sq);
      }
    }
  }
}

// ---------------------------------------------------------------------------
// G[b,c,d] = sum_l q1n[l,c] * k1n[l,d] where q1n/k1n = relu(LN_over_L)/sqrt(L)
// WMMA with K = l; each wave owns two of the sixteen 16x16 (c,d) tiles,
// interleaved in one l-loop (two independent WMMA chains, shared LN params).
// ---------------------------------------------------------------------------
__global__ void gram_kernel(const float* __restrict__ Q1b, const float* __restrict__ K1b,
                            const float* __restrict__ lnqg, const float* __restrict__ lnqb,
                            const float* __restrict__ lnkg, const float* __restrict__ lnkb,
                            const float* __restrict__ stats, float* __restrict__ G,
                            int chunk) {
  int b = blockIdx.y;
  int base = blockIdx.x * chunk;
  int wave = threadIdx.x >> 5;
  int lane = threadIdx.x & 31, hx = lane >> 4, lr = lane & 15;
  const float invL = 1.f / (float)LTOT;
  const float invSqrtL = rsqrtf((float)LTOT);
  const int B64 = BATCH * 64;
  int end = base + chunk;
  if (end > LTOT) end = LTOT;
  int tt0 = wave, tt1 = wave + 8;
  int ct0 = tt0 >> 2, dt0 = tt0 & 3, ct1 = tt1 >> 2, dt1 = tt1 & 3;
  int c0 = ct0 * 16 + lr, d0 = dt0 * 16 + lr;
  int c1 = ct1 * 16 + lr, d1 = dt1 * 16 + lr;
  float mq0 = stats[2 * B64 + b * 64 + c0] * invL;
  float rq0 = rsqrtf(stats[3 * B64 + b * 64 + c0] * invL - mq0 * mq0 + 1e-5f);
  float mk0 = stats[4 * B64 + b * 64 + d0] * invL;
  float rk0 = rsqrtf(stats[5 * B64 + b * 64 + d0] * invL - mk0 * mk0 + 1e-5f);
  float mq1 = stats[2 * B64 + b * 64 + c1] * invL;
  float rq1 = rsqrtf(stats[3 * B64 + b * 64 + c1] * invL - mq1 * mq1 + 1e-5f);
  float mk1 = stats[4 * B64 + b * 64 + d1] * invL;
  float rk1 = rsqrtf(stats[5 * B64 + b * 64 + d1] * invL - mk1 * mk1 + 1e-5f);
  size_t rowOff = ((size_t)b * LTOT + base + 2 * hx) * DIM;
  const float* qp0 = Q1b + rowOff + c0;
  const float* qp1 = Q1b + rowOff + c1;
  const float* kp0 = K1b + rowOff + d0;
  const float* kp1 = K1b + rowOff + d1;
  const float* gq = lnqg + base + 2 * hx;
  const float* gb = lnqb + base + 2 * hx;
  const float* kg = lnkg + base + 2 * hx;
  const float* kb = lnkb + base + 2 * hx;
  v8f acc0 = {}, acc1 = {};
  int n4 = (end - base) >> 2;  // LTOT % 4 == 0
  for (int i = 0; i < n4; i++) {
    v2f gqv = *(const v2f*)gq;
    v2f gbv = *(const v2f*)gb;
    v2f kgv = *(const v2f*)kg;
    v2f kbv = *(const v2f*)kb;
    gq += 4; gb += 4; kg += 4; kb += 4;
    v2f a0, b0, a1, b1;
    a0.x = fmaxf((qp0[0]  - mq0) * rq0 * gqv.x + gbv.x, 0.f) * invSqrtL;
    a0.y = fmaxf((qp0[64] - mq0) * rq0 * gqv.y + gbv.y, 0.f) * invSqrtL;
    b0.x = fmaxf((kp0[0]  - mk0) * rk0 * kgv.x + kbv.x, 0.f) * invSqrtL;
    b0.y = fmaxf((kp0[64] - mk0) * rk0 * kgv.y + kbv.y, 0.f) * invSqrtL;
    a1.x = fmaxf((qp1[0]  - mq1) * rq1 * gqv.x + gbv.x, 0.f) * invSqrtL;
    a1.y = fmaxf((qp1[64] - mq1) * rq1 * gqv.y + gbv.y, 0.f) * invSqrtL;
    b1.x = fmaxf((kp1[0]  - mk1) * rk1 * kgv.x + kbv.x, 0.f) * invSqrtL;
    b1.y = fmaxf((kp1[64] - mk1) * rk1 * kgv.y + kbv.y, 0.f) * invSqrtL;
    qp0 += 256; kp0 += 256; qp1 += 256; kp1 += 256;
    acc0 = wmma4(a0, b0, acc0);
    acc1 = wmma4(a1, b1, acc1);
  }
#pragma unroll
  for (int r = 0; r < 8; r++) {
    int cc0 = ct0 * 16 + r + 8 * hx;
    int cc1 = ct1 * 16 + r + 8 * hx;
    atomicAdd(&G[(size_t)b * 4096 + cc0 * 64 + dt0 * 16 + lr], acc0[r]);
    atomicAdd(&G[(size_t)b * 4096 + cc1 * 64 + dt1 * 16 + lr], acc1[r]);
  }
}

// Row-wise softmax over d: mid[b,c,:] = softmax(G[b,c,:]).  One wave per row.
__global__ void softmax_kernel(const float* __restrict__ G, float* __restrict__ mid) {
  int wid = blockIdx.x * (blockDim.x >> 5) + (threadIdx.x >> 5);
  if (wid >= BATCH * 64) return;
  int lane = threadIdx.x & 31;
  v2f v = *(const v2f*)(G + (size_t)wid * 64 + 2 * lane);
  float mx = fmaxf(v.x, v.y);
  for (int m = 16; m >= 1; m >>= 1) mx = fmaxf(mx, __shfl_xor(mx, m, 32));
  float e0 = expf(v.x - mx), e1 = expf(v.y - mx);
  float s = e0 + e1;
  for (int m = 16; m >= 1; m >>= 1) s += __shfl_xor(s, m, 32);
  float inv = 1.f / s;
  v2f o;
  o.x = e0 * inv;
  o.y = e1 * inv;
  *(v2f*)(mid + (size_t)wid * 64 + 2 * lane) = o;
}

// ---------------------------------------------------------------------------
// resmid[l,:] = (Q[l,:] * diag * Z[l]) @ mid[b];  Z computed in-register.
// ---------------------------------------------------------------------------
__global__ void resmid_kernel(const float* __restrict__ Qb, const float* __restrict__ stats,
                              const float* __restrict__ mid, float* __restrict__ res,
                              int tilesPerBatch) {
  int wid = blockIdx.x * (blockDim.x >> 5) + (threadIdx.x >> 5);
  int b = wid / tilesPerBatch;
  if (b >= BATCH) return;
  int tile = wid - b * tilesPerBatch;
  int tok0 = tile * 16;
  int lane = threadIdx.x & 31, hx = lane >> 4, lr = lane & 15;
  int tok = tok0 + lr;
  float mask = tok < LTOT ? 1.f : 0.f;
  int tokC = tok < LTOT ? tok : (LTOT - 1);
  const int B64 = BATCH * 64;
  const float* qrow = Qb + ((size_t)b * LTOT + tokC) * DIM + 2 * hx;
  const float* ks = stats + 0 * B64 + b * 64 + 2 * hx;
  const float* dg = stats + 1 * B64 + b * 64 + 2 * hx;
  v2f a[16];
  float zp = 0.f;
#pragma unroll
  for (int s = 0; s < 16; s++) {
    v2f qv = *(const v2f*)(qrow + 4 * s) * mask;
    v2f kv = *(const v2f*)(ks + 4 * s);
    v2f dv = *(const v2f*)(dg + 4 * s);
    zp += qv.x * (kv.x + 1e-6f) + qv.y * (kv.y + 1e-6f);
    a[s].x = qv.x * dv.x;
    a[s].y = qv.y * dv.y;
  }
  zp += __shfl_xor(zp, 16, 32);
  float Z = mask * (1.f / zp);
#pragma unroll
  for (int s = 0; s < 16; s++) a[s] = a[s] * Z;
  v8f acc[4];
  gemm64(a, mid + (size_t)b * 4096, hx, lr, acc);
#pragma unroll
  for (int t = 0; t < 4; t++) {
#pragma unroll
    for (int r = 0; r < 8; r++) {
      int m = r + 8 * hx;
      if ((tok0 + m) < LTOT)
        res[((size_t)b * LTOT + tok0 + m) * DIM + t * 16 + lr] = acc[t][r];
    }
  }
}

// u = t1 + gamma*res; a = LN(u, ln); t1 <- LN(t1 + a, norm1).  Wave per token.
__global__ void finalize_kernel(float* __restrict__ t1, const float* __restrict__ res,
                                const float* __restrict__ gamma_p,
                                const float* __restrict__ lng, const float* __restrict__ lnb,
                                const float* __restrict__ n1g, const float* __restrict__ n1b) {
  int wid = blockIdx.x * (blockDim.x >> 5) + (threadIdx.x >> 5);
  if (wid >= BATCH * LTOT) return;
  int lane = threadIdx.x & 31;
  float gamma = gamma_p[0];
  float* trow = t1 + (size_t)wid * DIM;
  int c0 = 2 * lane;
  v2f xv = *(const v2f*)(trow + c0);
  v2f rv = *(const v2f*)(res + (size_t)wid * DIM + c0);
  float u0 = xv.x + gamma * rv.x, u1 = xv.y + gamma * rv.y;
  float s = u0 + u1, sq = u0 * u0 + u1 * u1;
  for (int m = 16; m >= 1; m >>= 1) {
    s += __shfl_xor(s, m, 32);
    sq += __shfl_xor(sq, m, 32);
  }
  float mean = s * (1.f / 64.f);
  float rstd = rsqrtf(sq * (1.f / 64.f) - mean * mean + 1e-5f);
  v2f g1 = *(const v2f*)(lng + c0);
  v2f b1 = *(const v2f*)(lnb + c0);
  float v0 = xv.x + (u0 - mean) * rstd * g1.x + b1.x;
  float v1 = xv.y + (u1 - mean) * rstd * g1.y + b1.y;
  s = v0 + v1;
  sq = v0 * v0 + v1 * v1;
  for (int m = 16; m >= 1; m >>= 1) {
    s += __shfl_xor(s, m, 32);
    sq += __shfl_xor(sq, m, 32);
  }
  mean = s * (1.f / 64.f);
  rstd = rsqrtf(sq * (1.f / 64.f) - mean * mean + 1e-5f);
  v2f g2 = *(const v2f*)(n1g + c0);
  v2f b2 = *(const v2f*)(n1b + c0);
  v2f o;
  o.x = (v0 - mean) * rstd * g2.x + b2.x;
  o.y = (v1 - mean) * rstd * g2.y + b2.y;
  *(v2f*)(trow + c0) = o;
}

// ---------------------------------------------------------------------------
// convT + residual + BN + ReLU: out[b,co,pix] layout [B,Cout,hw2]
// ---------------------------------------------------------------------------
__global__ void out_kernel(const float* __restrict__ t1, const float* __restrict__ wT,
                           const float* __restrict__ bT, const float* __restrict__ x,
                           const float* __restrict__ bns, const float* __restrict__ bnb,
                           const float* __restrict__ bnm, const float* __restrict__ bnv,
                           const float* __restrict__ alpha_p, int alphaIdx,
                           float* __restrict__ out, int Cout, int hw2, int segStart,
                           int tilesPerBatch) {
  int wid = blockIdx.x * (blockDim.x >> 5) + (threadIdx.x >> 5);
  int b = wid / tilesPerBatch;
  if (b >= BATCH) return;
  int tile = wid - b * tilesPerBatch;
  int pixBase = tile * 16;
  int lane = threadIdx.x & 31, hx = lane >> 4, lr = lane & 15;
  int pix0 = pixBase + lr;
  float mask = pix0 < hw2 ? 1.f : 0.f;
  int pixC = pix0 < hw2 ? pix0 : (hw2 - 1);
  float alpha = alpha_p[alphaIdx];
  const float* trow = t1 + ((size_t)b * LTOT + segStart + pixC) * DIM + 2 * hx;
  v2f a[16];
#pragma unroll
  for (int s = 0; s < 16; s++) a[s] = *(const v2f*)(trow + 4 * s) * mask;
  int ntiles = Cout >> 4;
  for (int nt = 0; nt < ntiles; nt++) {
    int co = nt * 16 + lr;
    const float* wn = wT + (size_t)co * DIM + 2 * hx;
    v8f acc = {};
#pragma unroll
    for (int s = 0; s < 16; s++) {
      v2f bv = *(const v2f*)(wn + 4 * s);
      acc = wmma4(a[s], bv, acc);
    }
    float bc = bT[co];
    float scale = bns[co] * rsqrtf(bnv[co] + 1e-5f);
    float mm = bnm[co], sh = bnb[co];
    const float* xrow = x + ((size_t)b * Cout + co) * hw2;
    float* orow = out + ((size_t)b * Cout + co) * hw2;
#pragma unroll
    for (int r = 0; r < 8; r++) {
      int pix = pixBase + r + 8 * hx;
      if (pix < hw2) {
        float y = acc[r] + bc;
        float zz = xrow[pix] + alpha * y;
        orow[pix] = fmaxf((zz - mm) * scale + sh, 0.f);
      }
    }
  }
}

// ---------------------------------------------------------------------------
extern "C" void kernel_launch(void* const* d_in, const int* in_sizes, int n_in,
                              void* d_out, int out_size, void* d_ws, size_t ws_size,
                              hipStream_t stream) {
  auto f = [&](int i) { return (const float*)d_in[i]; };
  static const int segS[5] = {0, 50176, 62720, 65856, 66640};
  static const int hw[5] = {224, 112, 56, 28, 14};
  static const int ch[5] = {64, 128, 256, 512, 512};

  const size_t BL64 = (size_t)BATCH * LTOT * DIM;
  float* t1 = (float*)d_ws;
  float* Qb = t1 + BL64;
  float* Q1b = Qb + BL64;
  float* K1b = Q1b + BL64;
  float* stats = K1b + BL64;           // [6][B][64]
  float* G = stats + 6 * BATCH * 64;   // [B][64][64]
  float* mid = G + BATCH * 4096;       // [B][64][64]
  float* resb = Q1b;                   // reuse Q1 buffer after gram

  // zero stats + G
  {
    int n = 6 * BATCH * 64 + BATCH * 4096;
    zero_kernel<<<(n + 255) / 256, 256, 0, stream>>>(stats, n);
  }
  // embed (per scale): conv1x1 -> raw tokens
  for (int i = 0; i < 5; i++) {
    int hw2 = hw[i] * hw[i];
    int tpb = (hw2 + 15) / 16;
    int blocks = (BATCH * tpb + 7) / 8;
    embed_kernel<<<blocks, 256, 0, stream>>>(f(i), f(5 + 8 * i), f(6 + 8 * i), t1,
                                             ch[i], hw2, segS[i], tpb);
  }
  // LN(norm) over channels, in place
  {
    int blocks = (BATCH * LTOT + 7) / 8;
    ln_tokens_kernel<<<blocks, 256, 0, stream>>>(t1, f(45), f(46));
  }
  // fused projections + streaming stats
  const int TPB = (LTOT + 15) / 16;
  {
    int blocks = (BATCH * TPB + 7) / 8;
    proj_kernel<<<blocks, 256, 0, stream>>>(t1, f(50), f(51), f(52), f(53), f(54),
                                            f(55), f(56), f(57), f(58), f(59), f(60),
                                            Qb, Q1b, K1b, stats, TPB);
  }
  // gram G = Q1n^T K1n (K = L)
  {
    int chunk = 1024;
    dim3 grid((LTOT + chunk - 1) / chunk, BATCH);
    gram_kernel<<<grid, 256, 0, stream>>>(Q1b, K1b, f(61), f(62), f(63), f(64), stats,
                                          G, chunk);
  }
  // mid = softmax(G)
  {
    int blocks = (BATCH * 64 + 7) / 8;
    softmax_kernel<<<blocks, 256, 0, stream>>>(G, mid);
  }
  // resmid = (Q * diag * Z) @ mid
  {
    int blocks = (BATCH * TPB + 7) / 8;
    resmid_kernel<<<blocks, 256, 0, stream>>>(Qb, stats, mid, resb, TPB);
  }
  // residuals + two LayerNorms -> t1 final (in place)
  {
    int blocks = (BATCH * LTOT + 7) / 8;
    finalize_kernel<<<blocks, 256, 0, stream>>>(t1, resb, f(65), f(66), f(67), f(47),
                                                f(48));
  }
  // per-scale convT + residual + BN + ReLU
  float* outp = (float*)d_out;
  size_t off = 0;
  for (int i = 0; i < 5; i++) {
    int hw2 = hw[i] * hw[i];
    int tpb = (hw2 + 15) / 16;
    int blocks = (BATCH * tpb + 7) / 8;
    out_kernel<<<blocks, 256, 0, stream>>>(t1, f(7 + 8 * i), f(8 + 8 * i), f(i),
                                           f(9 + 8 * i), f(10 + 8 * i), f(11 + 8 * i),
                                           f(12 + 8 * i), f(49), i, outp + off, ch[i],
                                           hw2, segS[i], tpb);
    off += (size_t)BATCH * ch[i] * hw2;
  }
}